// GNN_Model_43885975831067
// MI455X (gfx1250) — compile-verified
//
#include <hip/hip_runtime.h>
#include <hip/hip_bf16.h>

// ---------------------------------------------------------------------------
// GNN pipeline for MI455X (gfx1250, wave32).
// Dead-code-eliminated: second GCN layer (agg_f / x_f layer 2) is never used
// by the reference output, so the dominant 100-dim scatter is skipped.
// All GEMM-shaped work uses V_WMMA_F32_16X16X4_F32 (bandwidth-bound -> f32).
// ---------------------------------------------------------------------------

typedef __attribute__((ext_vector_type(2))) float v2f;
typedef __attribute__((ext_vector_type(8))) float v8f;

__device__ __forceinline__ v8f wmma_k4(v2f a, v2f b, v8f c) {
  // 8-arg pattern (probe-confirmed for 16x16x{4,32}):
  // (neg_a, A, neg_b, B, c_mod, C, reuse_a, reuse_b)
  return __builtin_amdgcn_wmma_f32_16x16x4_f32(false, a, false, b, (short)0, c,
                                               false, false);
}

// ---------------- constants / workspace layout (float offsets) -------------
#define N_F0   8192
#define N_F    524288        // 8192*64
#define N_P    65536         // 16*4096
#define E_FULL 4194304
#define E_HALF 2097152
#define KD1    409400        // (4096-2)*100
#define NPAD   512           // dense1 N padded (500 -> 512)

#define OFF_XF   ((size_t)0)            // 1572864  x_f (524288 x 3)
#define OFF_ACC  ((size_t)1572864)      // 196608   scatter accumulators
#define OFF_CNT  ((size_t)1769472)      // 65536    scatter counts
#define OFF_XP   ((size_t)1835008)      // 6553600  x_p (65536 x 100) (+64 pad)
#define OFF_Y1   ((size_t)8388672)      // 6552000  conv1 out (16,100,4095) (+64 pad)
#define OFF_Y2   ((size_t)14940736)     // 6550400  conv2 out (16,100,4094) (+64 pad)
#define OFF_D1   ((size_t)21491200)     // 8192     dense1 acc (16 x 512)
// total 21499392 floats = ~86 MB

// ---------------------------------------------------------------------------
__global__ void zero_f32(float* __restrict__ p, int n) {
  int i = blockIdx.x * blockDim.x + threadIdx.x;
  if (i < n) p[i] = 0.f;
}

// ---------------------------------------------------------------------------
// Generic C[m,n] = opt_relu( sum_k A[m,k]*B[n,k] + bias[n] )
// A: MxK row-major, B: NxK row-major (i.e. GEMM-NT). One 16x16 tile per wave.
// ---------------------------------------------------------------------------
__global__ void wmma_gemm_nt(const float* __restrict__ A,
                             const float* __restrict__ B,
                             const float* __restrict__ bias,
                             float* __restrict__ C,
                             int M, int N, int K, int ldc, int tiles_n,
                             int relu_flag) {
  int w    = blockIdx.x * (blockDim.x >> 5) + (threadIdx.x >> 5);
  int lane = threadIdx.x & 31;
  int tiles_m = (M + 15) >> 4;
  if (w >= tiles_m * tiles_n) return;           // uniform per wave
  int nt = w % tiles_n, mt = w / tiles_n;
  int l15 = lane & 15, hi = lane >> 4;

  int arow = mt * 16 + l15; if (arow > M - 1) arow = M - 1;
  int brow = nt * 16 + l15; if (brow > N - 1) brow = N - 1;
  const float* ap = A + (size_t)arow * K + 2 * hi;   // A tile: V0={k,}, V1={k+1}
  const float* bp = B + (size_t)brow * K + 2 * hi;

  v8f c = {};
  #pragma unroll 8
  for (int k = 0; k < K; k += 4) {
    v2f a = *(const v2f*)(ap + k);
    v2f b = *(const v2f*)(bp + k);
    c = wmma_k4(a, b, c);
  }

  int n = nt * 16 + l15;
  if (n < N) {
    float bv = bias ? bias[n] : 0.f;
    int mbase = mt * 16 + hi * 8;
    #pragma unroll
    for (int r = 0; r < 8; ++r) {
      int m = mbase + r;
      if (m < M) {
        float v = c[r] + bv;
        if (relu_flag) v = fmaxf(v, 0.f);
        C[(size_t)m * ldc + n] = v;
      }
    }
  }
}

// ---------------------------------------------------------------------------
// dense1 (M=16 fixed): K split across waves, f32 atomic accumulation into
// Cacc (16 x NPAD). 32 N-tiles x 100 K-chunks = 3200 waves -> saturates HBM
// on the 820 MB W_d1 stream.
// ---------------------------------------------------------------------------
__global__ void wmma_gemm_split(const float* __restrict__ A,   // 16 x K
                                const float* __restrict__ B,   // N x K
                                float* __restrict__ Cacc,      // 16 x NPAD
                                int N, int K, int kchunk) {
  int w    = blockIdx.x * (blockDim.x >> 5) + (threadIdx.x >> 5);
  int lane = threadIdx.x & 31;
  int nt = w & 31;          // 32 n-tiles
  int kc = w >> 5;
  int k0 = kc * kchunk;
  int k1 = k0 + kchunk; if (k1 > K) k1 = K;
  int l15 = lane & 15, hi = lane >> 4;

  int brow = nt * 16 + l15; if (brow > N - 1) brow = N - 1;
  const float* ap = A + (size_t)l15 * K + 2 * hi;     // M row = lane&15
  const float* bp = B + (size_t)brow * K + 2 * hi;

  v8f c = {};
  #pragma unroll 8
  for (int k = k0; k < k1; k += 4) {
    v2f a = *(const v2f*)(ap + k);
    v2f b = *(const v2f*)(bp + k);
    c = wmma_k4(a, b, c);
  }

  int n = nt * 16 + l15;
  int mbase = hi * 8;
  #pragma unroll
  for (int r = 0; r < 8; ++r)
    atomicAdd(&Cacc[(size_t)(mbase + r) * NPAD + n], c[r]);
}

// ---------------------------------------------------------------------------
// 1-D conv (kernel width 2, 100->100 channels) as implicit WMMA GEMM.
// Y[b,o,s] = relu( sum_{i,t} W[o, i*2+t] * X[b, i, s+t] )
// K = 200, A = weights (row-major o x 200), B built on the fly.
// One 16(out-ch) x 16(pos) tile per wave.
// ---------------------------------------------------------------------------
__global__ void wmma_conv2tap(const float* __restrict__ X,
                              const float* __restrict__ W,
                              float* __restrict__ Y,
                              int in_row, int out_row,
                              long in_bstride, long out_bstride) {
  int w    = blockIdx.x * (blockDim.x >> 5) + (threadIdx.x >> 5);
  int lane = threadIdx.x & 31;
  int nt = w & 255;                 // 256 position tiles
  int mt = (w >> 8) % 7;            // 7 out-channel tiles (112 >= 100)
  int b  = w / (256 * 7);
  int l15 = lane & 15, hi = lane >> 4;
  int s0 = nt * 16;

  int o = mt * 16 + l15; if (o > 99) o = 99;
  const float* wp = W + o * 200 + 2 * hi;
  // B lane address: k = kb + hi*2 (+0/+1) -> i = kb/2 + hi, t = 0/1
  const float* xp = X + (size_t)b * in_bstride + (size_t)hi * in_row + s0 + l15;

  v8f c = {};
  #pragma unroll 2
  for (int kb = 0; kb < 200; kb += 4) {
    v2f a = *(const v2f*)(wp + kb);
    const float* xpi = xp + (size_t)(kb >> 1) * in_row;
    v2f bv; bv.x = xpi[0]; bv.y = xpi[1];   // taps t=0,1 (contiguous)
    c = wmma_k4(a, bv, c);
  }

  int s = s0 + l15;
  if (s < out_row) {
    int obase = mt * 16 + hi * 8;
    #pragma unroll
    for (int r = 0; r < 8; ++r) {
      int oo = obase + r;
      if (oo < 100)
        Y[(size_t)b * out_bstride + (size_t)oo * out_row + s] = fmaxf(c[r], 0.f);
    }
  }
}

// ---------------------------------------------------------------------------
// Edge scatter: mean-aggregate x_f rows into N_P nodes (forward edges only,
// edge_index[:, ::2]).
// ---------------------------------------------------------------------------
__global__ void scatter_edges(const int* __restrict__ ei,
                              const float* __restrict__ xf,
                              float* __restrict__ acc,
                              float* __restrict__ cnt) {
  int e = blockIdx.x * blockDim.x + threadIdx.x;
  if (e >= E_HALF) return;
  int src = ei[2 * (size_t)e];                   // row 0, even cols -> x_f row
  int dst = ei[(size_t)E_FULL + 2 * (size_t)e];  // row 1, even cols -> node
  const float* v = xf + (size_t)src * 3;
  atomicAdd(&acc[(size_t)dst * 3 + 0], v[0]);
  atomicAdd(&acc[(size_t)dst * 3 + 1], v[1]);
  atomicAdd(&acc[(size_t)dst * 3 + 2], v[2]);
  atomicAdd(&cnt[dst], 1.f);
}

// ---------------------------------------------------------------------------
// x_p = relu( mean(acc) @ W_l1.T + b_l1 + x_dst @ W_r1.T )   (65536 x 100)
// ---------------------------------------------------------------------------
__global__ void gcn_layer1(const float* __restrict__ acc,
                           const float* __restrict__ cnt,
                           const float* __restrict__ xdst,
                           const float* __restrict__ Wl1,
                           const float* __restrict__ bl1,
                           const float* __restrict__ Wr1,
                           float* __restrict__ xp) {
  int p = blockIdx.x;
  int c = threadIdx.x;
  float m  = fmaxf(cnt[p], 1.f);
  float a0 = acc[(size_t)p * 3 + 0] / m;
  float a1 = acc[(size_t)p * 3 + 1] / m;
  float a2 = acc[(size_t)p * 3 + 2] / m;
  float d0 = xdst[(size_t)p * 3 + 0];
  float d1 = xdst[(size_t)p * 3 + 1];
  float d2 = xdst[(size_t)p * 3 + 2];
  if (c < 100) {
    float v = a0 * Wl1[c * 3 + 0] + a1 * Wl1[c * 3 + 1] + a2 * Wl1[c * 3 + 2]
            + d0 * Wr1[c * 3 + 0] + d1 * Wr1[c * 3 + 1] + d2 * Wr1[c * 3 + 2]
            + bl1[c];
    xp[(size_t)p * 100 + c] = fmaxf(v, 0.f);
  }
}

// ---------------------------------------------------------------------------
// head: out = softmax( relu(z[:, :500]) @ W_d2.T )   z = dense1 acc (16x512)
// ---------------------------------------------------------------------------
__global__ void head_kernel(const float* __restrict__ z,
                            const float* __restrict__ Wd2,
                            float* __restrict__ out) {
  __shared__ float logits[107];
  __shared__ float red[2];
  int m = blockIdx.x, n = threadIdx.x;
  if (n < 107) {
    const float* zr = z + (size_t)m * NPAD;
    const float* wr = Wd2 + (size_t)n * 500;
    float s = 0.f;
    #pragma unroll 4
    for (int k = 0; k < 500; ++k) s += fmaxf(zr[k], 0.f) * wr[k];
    logits[n] = s;
  }
  __syncthreads();
  if (n == 0) {
    float mx = logits[0];
    for (int i = 1; i < 107; ++i) mx = fmaxf(mx, logits[i]);
    float su = 0.f;
    for (int i = 0; i < 107; ++i) su += __expf(logits[i] - mx);
    red[0] = mx; red[1] = su;
  }
  __syncthreads();
  if (n < 107) out[(size_t)m * 107 + n] = __expf(logits[n] - red[0]) / red[1];
}

// ---------------------------------------------------------------------------
extern "C" void kernel_launch(void* const* d_in, const int* in_sizes, int n_in,
                              void* d_out, int out_size, void* d_ws,
                              size_t ws_size, hipStream_t stream) {
  (void)in_sizes; (void)n_in; (void)out_size; (void)ws_size;
  const float* x_src = (const float*)d_in[0];
  const float* x_dst = (const float*)d_in[1];
  const int*   ei    = (const int*)d_in[2];
  const float* W_fd  = (const float*)d_in[3];
  const float* b_fd  = (const float*)d_in[4];
  const float* W_l1  = (const float*)d_in[5];
  const float* b_l1  = (const float*)d_in[6];
  const float* W_r1  = (const float*)d_in[7];
  // d_in[8..10] (W_l2, b_l2, W_r2) feed dead code in the reference -> unused
  const float* Wc1   = (const float*)d_in[11];
  const float* Wc2   = (const float*)d_in[12];
  const float* W_d1  = (const float*)d_in[13];
  const float* W_d2  = (const float*)d_in[14];

  float* ws  = (float*)d_ws;
  float* xf  = ws + OFF_XF;
  float* acc = ws + OFF_ACC;
  float* cnt = ws + OFF_CNT;
  float* xp  = ws + OFF_XP;
  float* y1  = ws + OFF_Y1;
  float* y2  = ws + OFF_Y2;
  float* d1  = ws + OFF_D1;

  // zero scatter accumulators (+counts, contiguous) and dense1 accumulators
  zero_f32<<<(N_P * 4 + 255) / 256, 256, 0, stream>>>(acc, N_P * 3 + N_P);
  zero_f32<<<(16 * NPAD + 255) / 256, 256, 0, stream>>>(d1, 16 * NPAD);

  // x_f = x_src @ W_fd.T + b_fd : M=8192, N=192, K=64 -> 512*12 = 6144 waves
  wmma_gemm_nt<<<768, 256, 0, stream>>>(x_src, W_fd, b_fd, xf,
                                        8192, 192, 64, 192, 12, 0);

  // segment mean aggregation over forward edges
  scatter_edges<<<E_HALF / 256, 256, 0, stream>>>(ei, xf, acc, cnt);

  // GCN layer 1 -> x_p (65536 x 100)
  gcn_layer1<<<N_P, 128, 0, stream>>>(acc, cnt, x_dst, W_l1, b_l1, W_r1, xp);

  // conv1: (16,100,4096) -> (16,100,4095); 16*7*256 = 28672 waves
  wmma_conv2tap<<<3584, 256, 0, stream>>>(xp, Wc1, y1, 4096, 4095,
                                          409600L, 409500L);
  // conv2: (16,100,4095) -> (16,100,4094)
  wmma_conv2tap<<<3584, 256, 0, stream>>>(y1, Wc2, y2, 4095, 4094,
                                          409500L, 409400L);

  // dense1: (16 x 409400) @ W_d1.T -> (16 x 500), K split 100 ways
  // 32 n-tiles * 100 k-chunks = 3200 waves -> 400 blocks of 8 waves
  wmma_gemm_split<<<400, 256, 0, stream>>>(y2, W_d1, d1, 500, KD1, 4096);

  // relu + dense2 + softmax
  head_kernel<<<16, 128, 0, stream>>>(d1, W_d2, (float*)d_out);
}